// Net_63101659513300
// MI455X (gfx1250) — compile-verified
//
#include <hip/hip_runtime.h>

typedef __attribute__((ext_vector_type(2))) float v2f;
typedef __attribute__((ext_vector_type(8))) float v8f;

#define NODE_DIM_IN 128
#define HID 16

// ---------------- degree / normalization ----------------

__global__ void init_deg_kernel(int* __restrict__ deg, int n) {
    int i = blockIdx.x * blockDim.x + threadIdx.x;
    if (i < n) deg[i] = 1;  // self loop
}

__global__ void count_deg_kernel(const int* __restrict__ dst, int* __restrict__ deg, int E) {
    int e = blockIdx.x * blockDim.x + threadIdx.x;
    if (e < E) atomicAdd(&deg[dst[e]], 1);
}

__global__ void dinv_kernel(const int* __restrict__ deg, float* __restrict__ dinv, int n) {
    int i = blockIdx.x * blockDim.x + threadIdx.x;
    if (i < n) dinv[i] = rsqrtf((float)deg[i]);
}

// ---------------- fused GEMM (WMMA f32 16x16x4) + per-row dinv scale ----------------
// G[m, 0..15] = dinv[m] * sum_k X[m,k] * W[k, 0..15]
// One wave per 16-row tile, grid-stride over tiles. B (W) fragments are loaded
// ONCE per wave into registers (K/4 * v2f) and reused for every tile.
// All branches are wave-uniform -> EXEC all-ones across every WMMA.
template <int K>
__global__ void gemm_scale_kernel(const float* __restrict__ X, const float* __restrict__ W,
                                  const float* __restrict__ dinv, float* __restrict__ G,
                                  int ntiles) {
    const int lane = threadIdx.x & 31;
    const int r  = lane & 15;       // A row within tile / B-N column
    const int hi = lane >> 4;       // K-pair select: lanes 0-15 -> K+0/1, 16-31 -> K+2/3

    // Hoisted B fragments: W[k][n], n = r, per-lane K pair.
    v2f b[K / 4];
#pragma unroll
    for (int k4 = 0; k4 < K / 4; ++k4) {
        int ka = k4 * 4 + 2 * hi;
        b[k4].x = W[ka * HID + r];
        b[k4].y = W[(ka + 1) * HID + r];
    }

    const int wave0  = blockIdx.x * (blockDim.x >> 5) + (threadIdx.x >> 5);
    const int nwaves = gridDim.x * (blockDim.x >> 5);

    for (int tile = wave0; tile < ntiles; tile += nwaves) {   // uniform per wave
        const long row0 = (long)tile * 16;
        const float* xrow = X + (row0 + r) * (long)K;
        v8f c = {};
#pragma unroll
        for (int k4 = 0; k4 < K / 4; ++k4) {
            int ka = k4 * 4 + 2 * hi;
            v2f a;
            a.x = xrow[ka];
            a.y = xrow[ka + 1];
            c = __builtin_amdgcn_wmma_f32_16x16x4_f32(
                    /*neg_a=*/false, a, /*neg_b=*/false, b[k4],
                    /*c_mod=*/(short)0, c, /*reuse_a=*/false, /*reuse_b=*/false);
        }
#pragma unroll
        for (int v = 0; v < 8; ++v) {
            long m = row0 + v + 8 * hi;     // C layout: lanes 0-15 -> M=v, 16-31 -> M=v+8
            G[m * HID + r] = c[v] * dinv[m];
        }
    }
}

// ---------------- edge scatter: agg[dst] += g[src] ----------------
// 16 lanes per edge (one per channel): coalesced gather + hardware f32 atomics.
__global__ void scatter_edges_kernel(const int* __restrict__ src, const int* __restrict__ dst,
                                     const float* __restrict__ g, float* __restrict__ agg, int E) {
    int t = blockIdx.x * blockDim.x + threadIdx.x;
    int e = t >> 4;
    int c = t & 15;
    if (e >= E) return;
    __builtin_prefetch(&src[e + 8192], 0, 1);   // global_prefetch_b8 on the index stream
    int s = src[e];
    int d = dst[e];
    float val = g[(long)s * HID + c];
    unsafeAtomicAdd(&agg[(long)d * HID + c], val);  // global_atomic_add_f32
}

// ---------------- layer-1 epilogue: z = relu(dinv*agg + b) ----------------
__global__ void relu_bias_kernel(const float* __restrict__ agg, const float* __restrict__ dinv,
                                 const float* __restrict__ b, float* __restrict__ z, int total) {
    int idx = blockIdx.x * blockDim.x + threadIdx.x;
    if (idx >= total) return;
    int c = idx & 15;
    int i = idx >> 4;
    float v = dinv[i] * agg[idx] + b[c];
    z[idx] = fmaxf(v, 0.0f);
}

// ---------------- layer-2 epilogue: out = log_softmax(dinv*agg + b) ----------------
__global__ void finalize_kernel(const float* __restrict__ agg, const float* __restrict__ dinv,
                                const float* __restrict__ b, float* __restrict__ out, int n) {
    int i = blockIdx.x * blockDim.x + threadIdx.x;
    if (i >= n) return;
    float di = dinv[i];
    float t[16];
    const float4* a4 = (const float4*)(agg + (long)i * HID);
    float mx = -3.0e38f;
#pragma unroll
    for (int q = 0; q < 4; ++q) {
        float4 v = a4[q];
        t[4 * q + 0] = di * v.x + b[4 * q + 0];
        t[4 * q + 1] = di * v.y + b[4 * q + 1];
        t[4 * q + 2] = di * v.z + b[4 * q + 2];
        t[4 * q + 3] = di * v.w + b[4 * q + 3];
    }
#pragma unroll
    for (int c = 0; c < 16; ++c) mx = fmaxf(mx, t[c]);
    float s = 0.0f;
#pragma unroll
    for (int c = 0; c < 16; ++c) s += expf(t[c] - mx);
    float ls = logf(s) + mx;
    float4* o4 = (float4*)(out + (long)i * HID);
#pragma unroll
    for (int q = 0; q < 4; ++q) {
        float4 v;
        v.x = t[4 * q + 0] - ls;
        v.y = t[4 * q + 1] - ls;
        v.z = t[4 * q + 2] - ls;
        v.w = t[4 * q + 3] - ls;
        o4[q] = v;
    }
}

extern "C" void kernel_launch(void* const* d_in, const int* in_sizes, int n_in,
                              void* d_out, int out_size, void* d_ws, size_t ws_size,
                              hipStream_t stream) {
    const float* x  = (const float*)d_in[0];
    const int*   ei = (const int*)d_in[1];
    const float* W1 = (const float*)d_in[2];
    const float* b1 = (const float*)d_in[3];
    const float* W2 = (const float*)d_in[4];
    const float* b2 = (const float*)d_in[5];
    float* out = (float*)d_out;

    const int N = in_sizes[0] / NODE_DIM_IN;   // 100000
    const int E = in_sizes[1] / 2;             // 3200000
    const int* srcv = ei;
    const int* dstv = ei + E;

    char* ws = (char*)d_ws;
    int*   deg  = (int*)ws;                                  // N ints
    float* dinv = (float*)(ws + (size_t)N * 4);              // N floats
    float* bufA = (float*)(ws + (size_t)N * 8);              // 16N floats (g)
    float* bufB = bufA + (size_t)N * HID;                    // 16N floats (agg)
    float* bufC = bufB + (size_t)N * HID;                    // 16N floats (z)
    const size_t featBytes = (size_t)N * HID * sizeof(float);

    // normalization
    init_deg_kernel<<<(N + 255) / 256, 256, 0, stream>>>(deg, N);
    count_deg_kernel<<<(E + 255) / 256, 256, 0, stream>>>(dstv, deg, E);
    dinv_kernel<<<(N + 255) / 256, 256, 0, stream>>>(deg, dinv, N);

    const int ntiles = (N + 15) / 16;          // 6250 (exact)
    const int gemmBlocks = 512;                // 4 waves/block (128 thr, wave32), grid-stride
    const int scatterBlocks = (E * 16 + 255) / 256;

    // layer 1: g1 = dinv * (x @ W1)
    gemm_scale_kernel<NODE_DIM_IN><<<gemmBlocks, 128, 0, stream>>>(x, W1, dinv, bufA, ntiles);
    // agg1 init = g1 (self-loop contribution)
    hipMemcpyAsync(bufB, bufA, featBytes, hipMemcpyDeviceToDevice, stream);
    scatter_edges_kernel<<<scatterBlocks, 256, 0, stream>>>(srcv, dstv, bufA, bufB, E);
    relu_bias_kernel<<<(N * HID + 255) / 256, 256, 0, stream>>>(bufB, dinv, b1, bufC, N * HID);

    // layer 2: g2 = dinv * (z @ W2)
    gemm_scale_kernel<HID><<<gemmBlocks, 128, 0, stream>>>(bufC, W2, dinv, bufA, ntiles);
    hipMemcpyAsync(bufB, bufA, featBytes, hipMemcpyDeviceToDevice, stream);
    scatter_edges_kernel<<<scatterBlocks, 256, 0, stream>>>(srcv, dstv, bufA, bufB, E);

    finalize_kernel<<<(N + 255) / 256, 256, 0, stream>>>(bufB, dinv, b2, out, N);
}